// GCN_r_26963804684657
// MI455X (gfx1250) — compile-verified
//
#include <hip/hip_runtime.h>

typedef float v2f __attribute__((ext_vector_type(2)));
typedef float v8f __attribute__((ext_vector_type(8)));

#define NNODES   100000
#define NEGSLOPE 0.01f

__device__ __forceinline__ float lrelu(float v) { return v > 0.0f ? v : NEGSLOPE * v; }

// ---------------- degree / norm ----------------
__global__ void k_deg_init(float* __restrict__ deg, int n) {
    int i = blockIdx.x * blockDim.x + threadIdx.x;
    if (i < n) deg[i] = 1.0f;  // self-loop
}

__global__ void k_deg_count(const int* __restrict__ edges, float* __restrict__ deg, int E) {
    int e = blockIdx.x * blockDim.x + threadIdx.x;
    if (e < E) atomicAdd(&deg[edges[E + e]], 1.0f);
}

__global__ void k_dinv(const float* __restrict__ deg, float* __restrict__ dinv, int n) {
    int i = blockIdx.x * blockDim.x + threadIdx.x;
    if (i < n) dinv[i] = rsqrtf(deg[i]);  // deg >= 1 always
}

// ---------------- Layer 1 GEMM: h1s = agg1 = dinv[m] * (x @ W1) ----------------
// fp32 WMMA 16x16x4; W1 (128x32 = 16KB) staged in LDS pre-swizzled as K-pairs:
// sW[(k/2)*64 + n*2 + (k&1)]  ->  B operand (W[k][n], W[k+1][n]) is one ds_load_b64.
__global__ void k_gemm1(const float* __restrict__ x, const float* __restrict__ W,
                        const float* __restrict__ dinv, float* __restrict__ hs,
                        float* __restrict__ agg, int n) {
    __shared__ float sW[128 * 32];
    for (int i = threadIdx.x; i < 128 * 32; i += blockDim.x) {
        const int k = i >> 5, c = i & 31;
        sW[(k >> 1) * 64 + c * 2 + (k & 1)] = W[i];
    }
    __syncthreads();

    const int wave = threadIdx.x >> 5;
    const int lane = threadIdx.x & 31;
    const int tile = blockIdx.x * (blockDim.x >> 5) + wave;
    const int rowBase = tile * 16;
    if (rowBase >= n) return;  // whole-wave exit; EXEC all-ones inside

    const int half = lane >> 4;      // 0: K=k..k+1, 1: K=k+2..k+3
    const int l16  = lane & 15;

    v8f c0 = {};
    v8f c1 = {};
    const float* xrow = x + (size_t)(rowBase + l16) * 128;

    #pragma unroll 4
    for (int k = 0; k < 128; k += 4) {
        const int ka = k + 2 * half;
        v2f a;  a.x = xrow[ka];  a.y = xrow[ka + 1];
        const int bidx = (ka >> 1) * 64 + l16 * 2;           // 8B-aligned pair
        v2f b0 = *(const v2f*)&sW[bidx];
        v2f b1 = *(const v2f*)&sW[bidx + 32];
        c0 = __builtin_amdgcn_wmma_f32_16x16x4_f32(false, a, false, b0, (short)0, c0, false, false);
        c1 = __builtin_amdgcn_wmma_f32_16x16x4_f32(false, a, false, b1, (short)0, c1, false, false);
    }

    #pragma unroll
    for (int r = 0; r < 8; ++r) {
        const int m = rowBase + r + 8 * half;  // C layout: VGPR r -> M=r / M=r+8
        const float s = dinv[m];
        const float v0 = c0[r] * s;
        const float v1 = c1[r] * s;
        hs [(size_t)m * 32 + l16]      = v0;   // message source values
        hs [(size_t)m * 32 + 16 + l16] = v1;
        agg[(size_t)m * 32 + l16]      = v0;   // self-loop term seeds accumulator
        agg[(size_t)m * 32 + 16 + l16] = v1;
    }
}

// ---------------- Layer 2 GEMM, fused with layer-1 epilogue ----------------
// A element = lrelu(dinv[row]*agg1[row][k] + b1[k]);  h2s = agg2 = dinv[m]*(h1 @ W2)
// W2 (32x16) staged in LDS with the same K-pair swizzle.
__global__ void k_gemm2(const float* __restrict__ agg1, const float* __restrict__ W,
                        const float* __restrict__ b1, const float* __restrict__ dinv,
                        float* __restrict__ hs, float* __restrict__ agg, int n) {
    __shared__ float sW[32 * 16];
    __shared__ float sB[32];
    for (int i = threadIdx.x; i < 32 * 16; i += blockDim.x) {
        const int k = i >> 4, c = i & 15;
        sW[(k >> 1) * 32 + c * 2 + (k & 1)] = W[i];
    }
    if (threadIdx.x < 32) sB[threadIdx.x] = b1[threadIdx.x];
    __syncthreads();

    const int wave = threadIdx.x >> 5;
    const int lane = threadIdx.x & 31;
    const int tile = blockIdx.x * (blockDim.x >> 5) + wave;
    const int rowBase = tile * 16;
    if (rowBase >= n) return;

    const int half = lane >> 4;
    const int l16  = lane & 15;

    v8f c = {};
    const int row = rowBase + l16;
    const float dvRow = dinv[row];
    const float* arow = agg1 + (size_t)row * 32;

    #pragma unroll
    for (int k = 0; k < 32; k += 4) {
        const int ka = k + 2 * half;
        v2f a;
        a.x = lrelu(dvRow * arow[ka]     + sB[ka]);      // fused layer-1 finish
        a.y = lrelu(dvRow * arow[ka + 1] + sB[ka + 1]);
        v2f b = *(const v2f*)&sW[(ka >> 1) * 32 + l16 * 2];
        c = __builtin_amdgcn_wmma_f32_16x16x4_f32(false, a, false, b, (short)0, c, false, false);
    }

    #pragma unroll
    for (int r = 0; r < 8; ++r) {
        const int m = rowBase + r + 8 * half;
        const float v = c[r] * dinv[m];
        hs [(size_t)m * 16 + l16] = v;
        agg[(size_t)m * 16 + l16] = v;   // self-loop seed
    }
}

// ---------------- edge aggregation ----------------
// agg[dst][j] += hs[src][j]  -- thread per (edge, col); lanes cover cols -> coalesced
__global__ void k_scatter(const int* __restrict__ edges, const float* __restrict__ hs,
                          float* __restrict__ agg, int E, int clog) {
    int idx = blockIdx.x * blockDim.x + threadIdx.x;
    int total = E << clog;
    if (idx < total) {
        int e = idx >> clog;
        int j = idx & ((1 << clog) - 1);
        int s = edges[e];
        int d = edges[E + e];
        atomicAdd(&agg[((size_t)d << clog) + j], hs[((size_t)s << clog) + j]);
    }
}

// ---------------- fused layer-2 finish + output projection ----------------
// out[v] = bfc + sum_j lrelu(dinv[v]*agg2[v][j] + b2[j]) * Wfc[j]
__global__ void k_out(const float* __restrict__ agg2, const float* __restrict__ dinv,
                      const float* __restrict__ b2, const float* __restrict__ Wfc,
                      const float* __restrict__ bfc, float* __restrict__ out, int n) {
    int v = blockIdx.x * blockDim.x + threadIdx.x;
    if (v < n) {
        const float dv = dinv[v];
        const float* row = agg2 + (size_t)v * 16;
        float s = bfc[0];
        #pragma unroll
        for (int j = 0; j < 16; ++j) s += lrelu(dv * row[j] + b2[j]) * Wfc[j];
        out[v] = s;
    }
}

extern "C" void kernel_launch(void* const* d_in, const int* in_sizes, int n_in,
                              void* d_out, int out_size, void* d_ws, size_t ws_size,
                              hipStream_t stream) {
    const float* x    = (const float*)d_in[0];
    const int*   ei   = (const int*)  d_in[1];   // [2, E] flat
    const float* W1   = (const float*)d_in[2];
    const float* b1   = (const float*)d_in[3];
    const float* W2   = (const float*)d_in[4];
    const float* b2   = (const float*)d_in[5];
    const float* Wfc  = (const float*)d_in[6];
    const float* bfc  = (const float*)d_in[7];
    float*       out  = (float*)d_out;

    const int n = NNODES;
    const int E = in_sizes[1] / 2;

    // workspace carve-up (512B aligned)
    char* ws = (char*)d_ws;
    size_t off = 0;
    auto carve = [&](size_t bytes) { char* p = ws + off; off = (off + bytes + 511) & ~(size_t)511; return p; };
    float* deg  = (float*)carve((size_t)n * 4);
    float* dinv = (float*)carve((size_t)n * 4);
    float* h1s  = (float*)carve((size_t)n * 32 * 4);
    float* agg1 = (float*)carve((size_t)n * 32 * 4);
    float* h2s  = (float*)carve((size_t)n * 16 * 4);
    float* agg2 = (float*)carve((size_t)n * 16 * 4);
    (void)ws_size;

    const int T = 256;
    const int wavesPerBlock = T / 32;
    const int tiles = (n + 15) / 16;
    const int gemmBlocks = (tiles + wavesPerBlock - 1) / wavesPerBlock;

    // symmetric normalization coefficients
    k_deg_init <<<(n + T - 1) / T, T, 0, stream>>>(deg, n);
    k_deg_count<<<(E + T - 1) / T, T, 0, stream>>>(ei, deg, E);
    k_dinv     <<<(n + T - 1) / T, T, 0, stream>>>(deg, dinv, n);

    // layer 1: WMMA GEMM (writes hs + self-seeded agg) then edge scatter
    k_gemm1  <<<gemmBlocks, T, 0, stream>>>(x, W1, dinv, h1s, agg1, n);
    k_scatter<<<((E << 5) + T - 1) / T, T, 0, stream>>>(ei, h1s, agg1, E, 5);

    // layer 2: finish(layer1) fused into WMMA GEMM A-load, then edge scatter
    k_gemm2  <<<gemmBlocks, T, 0, stream>>>(agg1, W2, b1, dinv, h2s, agg2, n);
    k_scatter<<<((E << 4) + T - 1) / T, T, 0, stream>>>(ei, h2s, agg2, E, 4);

    // finish(layer2) fused with final 16->1 projection
    k_out<<<(n + T - 1) / T, T, 0, stream>>>(agg2, dinv, b2, Wfc, bfc, out, n);
}